// LSTMModule_21131239096794
// MI455X (gfx1250) — compile-verified
//
#include <hip/hip_runtime.h>
#include <math.h>

// Problem constants (reference: B, T, F, U = 64, 512, 1024, 256)
#define B_   64
#define T_   512
#define F_   1024
#define U_   256
#define G4_  1024        // 4*U (i,f,g,o gate columns)

typedef __attribute__((ext_vector_type(2))) float v2f;
typedef __attribute__((ext_vector_type(8))) float v8f;

// CDNA5 fp32 WMMA: D(16x16) = A(16x4) * B(4x16) + C
__device__ __forceinline__ v8f wmma_f32_k4(v2f a, v2f b, v8f c) {
  return __builtin_amdgcn_wmma_f32_16x16x4_f32(false, a, false, b, (short)0, c,
                                               false, false);
}

__device__ __forceinline__ float sigf(float x) {
  return 1.0f / (1.0f + __expf(-x));
}

// ---------------------------------------------------------------------------
// mask[b*T+t] = any(x[b,t,:] != 0) ? 1 : 0   (one 256-thread block per row)
// ---------------------------------------------------------------------------
__global__ __launch_bounds__(256) void row_mask(const float* __restrict__ X,
                                                float* __restrict__ maskf) {
  __shared__ int red[8];
  const int row = blockIdx.x;
  const float* xr = X + (size_t)row * F_;
  int nz = 0;
  for (int i = threadIdx.x; i < F_; i += 256) nz |= (xr[i] != 0.0f) ? 1 : 0;
  for (int off = 16; off > 0; off >>= 1) nz |= __shfl_down(nz, off, 32);
  if ((threadIdx.x & 31) == 0) red[threadIdx.x >> 5] = nz;
  __syncthreads();
  if (threadIdx.x == 0) {
    int a = 0;
#pragma unroll
    for (int w = 0; w < 8; ++w) a |= red[w];
    maskf[row] = a ? 1.0f : 0.0f;
  }
}

__global__ void zero_f32(float* __restrict__ p, int n) {
  int i = blockIdx.x * blockDim.x + threadIdx.x;
  if (i < n) p[i] = 0.0f;
}

// ---------------------------------------------------------------------------
// Z = relu(X @ W_fc + b_fc):  M=32768, K=1024, N=256, fp32 WMMA.
// Wave tile: 32x64 -> acc[2][4]; every B fragment feeds 2 WMMAs, so the
// steady-state loop is (2x b64 A + 8x b32 B) per 8 WMMAs (~1.25 VMEM/WMMA).
// Grid: 1024 m-tiles(32) * 4 n-groups = 4096 waves = 512 blocks @ 256.
// ---------------------------------------------------------------------------
__global__ __launch_bounds__(256) void fc_relu_wmma(const float* __restrict__ X,
                                                    const float* __restrict__ W,
                                                    const float* __restrict__ bias,
                                                    float* __restrict__ Z) {
  const int lane = threadIdx.x & 31;
  const int gw   = blockIdx.x * 8 + (threadIdx.x >> 5);
  const int mt   = gw >> 2;
  const int ng   = gw & 3;
  const int half = lane >> 4;       // lanes 16-31 carry K = k+2,k+3 (ISA 7.12.2)
  const int lr   = lane & 15;
  const int m0 = mt * 32, n0 = ng * 64;

  const v8f vzero = {0.f, 0.f, 0.f, 0.f, 0.f, 0.f, 0.f, 0.f};
  v8f acc[2][4] = {{vzero, vzero, vzero, vzero}, {vzero, vzero, vzero, vzero}};

  const float* arow0 = X + (size_t)(m0 + lr) * F_ + 2 * half;
  const float* arow1 = arow0 + (size_t)16 * F_;
  const float* bbase = W + (size_t)(2 * half) * U_;

#pragma unroll 4
  for (int kk = 0; kk < F_; kk += 4) {
    if ((kk & 127) == 0) {
      __builtin_prefetch(arow0 + kk + 256, 0, 0);
      __builtin_prefetch(arow1 + kk + 256, 0, 0);
    }
    v2f a0; a0.x = arow0[kk]; a0.y = arow0[kk + 1];
    v2f a1; a1.x = arow1[kk]; a1.y = arow1[kk + 1];
#pragma unroll
    for (int s = 0; s < 4; ++s) {
      const int n = n0 + s * 16 + lr;
      v2f b; b.x = bbase[(size_t)kk * U_ + n];
             b.y = bbase[(size_t)kk * U_ + U_ + n];
      acc[0][s] = wmma_f32_k4(a0, b, acc[0][s]);
      acc[1][s] = wmma_f32_k4(a1, b, acc[1][s]);
    }
  }
#pragma unroll
  for (int mi = 0; mi < 2; ++mi) {
#pragma unroll
    for (int s = 0; s < 4; ++s) {
      const int n = n0 + s * 16 + lr;
#pragma unroll
      for (int r = 0; r < 8; ++r) {
        const int row = m0 + mi * 16 + r + 8 * half;  // C/D: VGPR r -> M=r, r+8
        float v = acc[mi][s][r] + bias[n];
        Z[(size_t)row * U_ + n] = v > 0.0f ? v : 0.0f;
      }
    }
  }
}

// ---------------------------------------------------------------------------
// G = Z @ W_x + b_lstm:  M=32768, K=256, N=1024.  Wave tile 32x64 (acc[2][4]).
// Output stored T-MAJOR: row (t*B + b), so each LSTM step reads a contiguous
// 64x1024 block (L2-friendly).
// Grid: 1024 m-tiles(32) * 16 n-groups = 16384 waves = 2048 blocks @ 256.
// ---------------------------------------------------------------------------
__global__ __launch_bounds__(256) void gates_wmma(const float* __restrict__ Z,
                                                  const float* __restrict__ Wx,
                                                  const float* __restrict__ bias,
                                                  float* __restrict__ G) {
  const int lane = threadIdx.x & 31;
  const int gw   = blockIdx.x * 8 + (threadIdx.x >> 5);
  const int mt   = gw >> 4;
  const int ng   = gw & 15;
  const int half = lane >> 4;
  const int lr   = lane & 15;
  const int m0 = mt * 32, n0 = ng * 64;

  const v8f vzero = {0.f, 0.f, 0.f, 0.f, 0.f, 0.f, 0.f, 0.f};
  v8f acc[2][4] = {{vzero, vzero, vzero, vzero}, {vzero, vzero, vzero, vzero}};

  const float* arow0 = Z + (size_t)(m0 + lr) * U_ + 2 * half;
  const float* arow1 = arow0 + (size_t)16 * U_;
  const float* bbase = Wx + (size_t)(2 * half) * G4_;

#pragma unroll 4
  for (int kk = 0; kk < U_; kk += 4) {
    v2f a0; a0.x = arow0[kk]; a0.y = arow0[kk + 1];
    v2f a1; a1.x = arow1[kk]; a1.y = arow1[kk + 1];
#pragma unroll
    for (int s = 0; s < 4; ++s) {
      const int n = n0 + s * 16 + lr;
      v2f b; b.x = bbase[(size_t)kk * G4_ + n];
             b.y = bbase[(size_t)kk * G4_ + G4_ + n];
      acc[0][s] = wmma_f32_k4(a0, b, acc[0][s]);
      acc[1][s] = wmma_f32_k4(a1, b, acc[1][s]);
    }
  }
#pragma unroll
  for (int mi = 0; mi < 2; ++mi) {
#pragma unroll
    for (int s = 0; s < 4; ++s) {
      const int n = n0 + s * 16 + lr;
#pragma unroll
      for (int r = 0; r < 8; ++r) {
        const int row = m0 + mi * 16 + r + 8 * half;  // row = b*T + t
        const int b   = row >> 9;                     // /T_ (512)
        const int t   = row & (T_ - 1);
        G[((size_t)t * B_ + b) * G4_ + n] = acc[mi][s][r] + bias[n];
      }
    }
  }
}

// ---------------------------------------------------------------------------
// One LSTM timestep. 64 waves: wave owns a 16(batch) x 16(units) patch and
// computes its FOUR gate columns (u, u+256, u+512, u+768) as 4 WMMA
// accumulators initialized from G, so the nonlinearity + state update are
// fully in-register. h/c ping-pong between launches (no grid sync needed).
// Grid: 8 blocks @ 256 (64 full waves, no divergence -> EXEC all ones).
// ---------------------------------------------------------------------------
__global__ __launch_bounds__(256) void lstm_step_wmma(
    const float* __restrict__ G, const float* __restrict__ Wh,
    const float* __restrict__ maskf,
    const float* __restrict__ h_in, const float* __restrict__ c_in,
    float* __restrict__ h_out, float* __restrict__ c_out,
    float* __restrict__ hs, int t) {
  const int lane = threadIdx.x & 31;
  const int gw   = blockIdx.x * 8 + (threadIdx.x >> 5);  // 0..63
  const int mt   = gw >> 4;                              // 4 batch tiles
  const int ut   = gw & 15;                              // 16 unit tiles
  const int half = lane >> 4;
  const int lr   = lane & 15;
  const int m0   = mt * 16;

  const float* Gt = G + (size_t)t * B_ * G4_;
  v8f acc[4];
#pragma unroll
  for (int g = 0; g < 4; ++g) {
    const int n = g * U_ + ut * 16 + lr;
#pragma unroll
    for (int r = 0; r < 8; ++r)
      acc[g][r] = Gt[(size_t)(m0 + r + 8 * half) * G4_ + n];
  }

  const float* arow  = h_in + (size_t)(m0 + lr) * U_ + 2 * half;
  const float* bbase = Wh + (size_t)(2 * half) * G4_;

#pragma unroll 4
  for (int kk = 0; kk < U_; kk += 4) {
    v2f a; a.x = arow[kk]; a.y = arow[kk + 1];
#pragma unroll
    for (int g = 0; g < 4; ++g) {
      const int n = g * U_ + ut * 16 + lr;
      v2f b; b.x = bbase[(size_t)kk * G4_ + n];
             b.y = bbase[(size_t)kk * G4_ + G4_ + n];
      acc[g] = wmma_f32_k4(a, b, acc[g]);
    }
  }

#pragma unroll
  for (int r = 0; r < 8; ++r) {
    const int brow = m0 + r + 8 * half;
    const int u    = ut * 16 + lr;
    const int hi   = brow * U_ + u;
    const float iv = sigf(acc[0][r]);
    const float fv = sigf(acc[1][r]);
    const float gv = tanhf(acc[2][r]);
    const float ov = sigf(acc[3][r]);
    const float co = c_in[hi], ho = h_in[hi];
    const float cn = fv * co + iv * gv;
    const float hn = ov * tanhf(cn);
    const float m  = maskf[brow * T_ + t];
    const float h2 = (m != 0.0f) ? hn : ho;
    const float c2 = (m != 0.0f) ? cn : co;
    h_out[hi] = h2;
    c_out[hi] = c2;
    hs[(size_t)(brow * T_ + t) * U_ + u] = h2;
  }
}

// ---------------------------------------------------------------------------
// out[b*T+t] = sigmoid(hs[b,t,:] . W_p + b_p); one wave per output row.
// ---------------------------------------------------------------------------
__global__ __launch_bounds__(256) void proj_sigmoid(const float* __restrict__ HS,
                                                    const float* __restrict__ Wp,
                                                    const float* __restrict__ bp,
                                                    float* __restrict__ out) {
  const int lane = threadIdx.x & 31;
  const int row  = blockIdx.x * 8 + (threadIdx.x >> 5);
  const float* hr = HS + (size_t)row * U_;
  float s = 0.0f;
#pragma unroll
  for (int u = lane; u < U_; u += 32) s += hr[u] * Wp[u];
  for (int off = 16; off > 0; off >>= 1) s += __shfl_down(s, off, 32);
  if (lane == 0) out[row] = sigf(s + bp[0]);
}

// ---------------------------------------------------------------------------
extern "C" void kernel_launch(void* const* d_in, const int* in_sizes, int n_in,
                              void* d_out, int out_size, void* d_ws, size_t ws_size,
                              hipStream_t stream) {
  (void)in_sizes; (void)n_in; (void)out_size; (void)ws_size;
  const float* x      = (const float*)d_in[0];
  const float* W_fc   = (const float*)d_in[1];
  const float* b_fc   = (const float*)d_in[2];
  const float* W_x    = (const float*)d_in[3];
  const float* W_h    = (const float*)d_in[4];
  const float* b_lstm = (const float*)d_in[5];
  const float* W_p    = (const float*)d_in[6];
  const float* b_p    = (const float*)d_in[7];
  float* out = (float*)d_out;

  // Workspace layout (floats). Total ~168 MB.
  float* ws    = (float*)d_ws;
  float* maskf = ws;                                  // B*T         = 32768
  float* zbuf  = maskf + (size_t)B_ * T_;             // B*T*U (z; later reused as hs)
  float* G     = zbuf + (size_t)B_ * T_ * U_;         // B*T*4U, t-major rows
  float* hbuf  = G + (size_t)B_ * T_ * G4_;           // 2 * B*U ping-pong
  float* cbuf  = hbuf + 2 * (size_t)B_ * U_;          // 2 * B*U ping-pong

  row_mask<<<B_ * T_, 256, 0, stream>>>(x, maskf);
  zero_f32<<<(4 * B_ * U_ + 255) / 256, 256, 0, stream>>>(hbuf, 4 * B_ * U_);

  fc_relu_wmma<<<512, 256, 0, stream>>>(x, W_fc, b_fc, zbuf);
  gates_wmma<<<2048, 256, 0, stream>>>(zbuf, W_x, b_lstm, G);

  float* hs = zbuf;  // z fully consumed by gates_wmma; reuse as hs (B*T*U)
  for (int t = 0; t < T_; ++t) {
    const float* h_in = hbuf + (size_t)(t & 1) * (B_ * U_);
    float* h_out      = hbuf + (size_t)((t + 1) & 1) * (B_ * U_);
    const float* c_in = cbuf + (size_t)(t & 1) * (B_ * U_);
    float* c_out      = cbuf + (size_t)((t + 1) & 1) * (B_ * U_);
    lstm_step_wmma<<<8, 256, 0, stream>>>(G, W_h, maskf, h_in, c_in,
                                          h_out, c_out, hs, t);
  }

  proj_sigmoid<<<4096, 256, 0, stream>>>(hs, W_p, b_p, out);
}